// GraphSMOTE_5428838662698
// MI455X (gfx1250) — compile-verified
//
#include <hip/hip_runtime.h>
#include <hip/hip_bf16.h>

typedef __attribute__((ext_vector_type(2))) float v2f;
typedef __attribute__((ext_vector_type(4))) float v4f;
typedef __attribute__((ext_vector_type(8))) float v8f;

#define T_N   8192
#define D_N   512
#define NPER_ 4
#define KNN   5           // neighbors kept (K-1 = 6-1)
#define LDSA_STRIDE 516   // 512 + 4 pad -> conflict-free WMMA fragment reads
#define TILE_STRIDE 20    // 16 + 4 pad -> 16B-aligned rows, conflict-free b128 reads

// ---------------------------------------------------------------------------
// Kernel A: row squared norms  sq[i] = sum_k X[i,k]^2
// ---------------------------------------------------------------------------
__global__ void smote_norms(const float* __restrict__ X, float* __restrict__ sq) {
    int row  = blockIdx.x * 8 + (threadIdx.x >> 5);
    int lane = threadIdx.x & 31;
    const float* xr = X + (size_t)row * D_N;
    float acc = 0.f;
    #pragma unroll
    for (int k = lane; k < D_N; k += 32) { float v = xr[k]; acc += v * v; }
    #pragma unroll
    for (int off = 16; off > 0; off >>= 1) acc += __shfl_xor(acc, off, 32);
    if (lane == 0) sq[row] = acc;
}

// branchless sorted-insert of (sv, jv) into ascending (bd[], bj[]) top-K
__device__ __forceinline__ void topk_insert(float (&bd)[KNN], int (&bj)[KNN],
                                            float sv, int jv) {
    #pragma unroll
    for (int s = 0; s < KNN; ++s) {
        float obd = bd[s];
        int   obj = bj[s];
        bool better = (sv < obd) || (sv == obd && jv < obj);
        bd[s] = better ? sv : obd;
        bj[s] = better ? jv : obj;
        sv    = better ? obd : sv;
        jv    = better ? obj : jv;
    }
}

// ---------------------------------------------------------------------------
// Kernel B: fused Gram (fp32 WMMA 16x16x4) + per-row top-5 selection.
// One workgroup (8 wave32) per 16-row block. Score surrogate per row i:
//   s(j) = sq[j] - 2 * (x_i . x_j)   (same ordering as euclidean distance)
// Selection uses all 32 lanes: lane m scans cols 0..7, lane m+16 scans 8..15.
// ---------------------------------------------------------------------------
__global__ __launch_bounds__(256, 1)
void smote_knn(const float* __restrict__ X, const float* __restrict__ sq,
               int* __restrict__ nnIdx) {
    __shared__ __align__(16) float ldsA[16 * LDSA_STRIDE];   // 33 KB A panel
    __shared__ __align__(16) float tile[8][16 * TILE_STRIDE];// per-wave 16x16 spill
    __shared__ float mD[8 * 32 * KNN];                       // per-half-lane top-5 d
    __shared__ int   mJ[8 * 32 * KNN];                       // per-half-lane top-5 j

    const int tid  = threadIdx.x;
    const int wave = tid >> 5;
    const int lane = tid & 31;
    const int i0   = blockIdx.x * 16;

    // stage A panel (16 x 512 fp32) into LDS with padded row stride (float4 I/O)
    for (int idx = tid; idx < 16 * (D_N / 4); idx += 256) {
        int r  = idx >> 7;            // 128 float4 per row
        int c4 = idx & 127;
        *(v4f*)(&ldsA[r * LDSA_STRIDE + c4 * 4]) =
            *(const v4f*)(X + (size_t)(i0 + r) * D_N + c4 * 4);
    }
    __syncthreads();

    // WMMA fp32 16x16x4 fragment addressing (ISA 7.12.2):
    //  A: lane L holds M = L&15, K = 2*(L>>4) + {0,1}   -> contiguous float2
    //  B: lane L holds N = L&15, K = 2*(L>>4) + {0,1}   -> same pattern (Gram)
    const int   mrow  = lane & 15;
    const int   half  = lane >> 4;
    const int   koff  = half << 1;
    const int   nbase = half << 3;       // selection column base: 0 or 8
    const float* aBase = &ldsA[mrow * LDSA_STRIDE + koff];

    float bd[KNN]; int bj[KNN];
    #pragma unroll
    for (int s = 0; s < KNN; ++s) { bd[s] = __builtin_inff(); bj[s] = 0x7fffffff; }

    for (int jt = wave; jt < T_N / 16; jt += 8) {
        const int j0 = jt * 16;
        const float* bBase = X + (size_t)(j0 + mrow) * D_N + koff;
        if (jt + 8 < T_N / 16)  // hint next tile's B panel toward L2/L0
            __builtin_prefetch(X + (size_t)((jt + 8) * 16 + mrow) * D_N + koff, 0, 1);

        // sq[j0 + n]: lane n (0..15) holds value for column n; broadcast via shfl
        float sqv = sq[j0 + mrow];

        v8f acc = {};
        #pragma unroll 8
        for (int k = 0; k < D_N; k += 4) {
            v2f a = *(const v2f*)(aBase + k);
            v2f b = *(const v2f*)(bBase + k);
            acc = __builtin_amdgcn_wmma_f32_16x16x4_f32(
                      false, a, false, b, (short)0, acc, false, false);
        }

        // spill accumulator: lane holds column n=mrow, rows m = v + 8*half
        float* tw = tile[wave];
        #pragma unroll
        for (int v = 0; v < 8; ++v)
            tw[(v + (half << 3)) * TILE_STRIDE + mrow] = acc[v];

        __builtin_amdgcn_wave_barrier();
        asm volatile("s_wait_dscnt 0x0" ::: "memory");

        // per-row top-5 update: both halves active, 8 columns each, b128 reads
        {
            const int i = i0 + mrow;
            const v4f* trow = (const v4f*)&tw[mrow * TILE_STRIDE + nbase];
            v4f t0 = trow[0];
            v4f t1 = trow[1];
            #pragma unroll
            for (int c = 0; c < 8; ++c) {
                const int n  = nbase + c;
                const int jv = j0 + n;
                float g2  = (c < 4) ? t0[c] : t1[c - 4];
                float sqn = __shfl(sqv, n, 32);
                float sv  = (jv == i) ? __builtin_inff() : fmaf(-2.0f, g2, sqn);
                // quick reject against current 5th-best (usually all lanes fail)
                bool enter = (sv < bd[KNN - 1]) ||
                             (sv == bd[KNN - 1] && jv < bj[KNN - 1]);
                if (enter) topk_insert(bd, bj, sv, jv);
            }
        }
        __builtin_amdgcn_wave_barrier();
    }

    // publish per-half-lane candidates
    #pragma unroll
    for (int s = 0; s < KNN; ++s) {
        mD[(wave * 32 + lane) * KNN + s] = bd[s];
        mJ[(wave * 32 + lane) * KNN + s] = bj[s];
    }
    __syncthreads();

    // merge the 8 waves x 2 halves per row; threads 0..15 own one row each
    if (tid < 16) {
        float fd[KNN]; int fj[KNN];
        #pragma unroll
        for (int s = 0; s < KNN; ++s) { fd[s] = __builtin_inff(); fj[s] = 0x7fffffff; }
        for (int w = 0; w < 8; ++w) {
            #pragma unroll
            for (int h = 0; h < 2; ++h) {
                #pragma unroll
                for (int s = 0; s < KNN; ++s) {
                    int base = (w * 32 + h * 16 + tid) * KNN + s;
                    topk_insert(fd, fj, mD[base], mJ[base]);
                }
            }
        }
        #pragma unroll
        for (int s = 0; s < KNN; ++s) nnIdx[(size_t)(i0 + tid) * KNN + s] = fj[s];
    }
}

// ---------------------------------------------------------------------------
// Kernel C: populate — syn = x_t + gap * (x_nn - x_t), float4 I/O
// one block per output row (32768 rows x 512)
// ---------------------------------------------------------------------------
__global__ void smote_populate(const float* __restrict__ X,
                               const float* __restrict__ gaps,
                               const int*   __restrict__ choice,
                               const int*   __restrict__ nnIdx,
                               float*       __restrict__ out) {
    int r = blockIdx.x;                 // 0 .. T*NPER-1
    int t = r >> 2, n = r & 3;
    float g  = gaps[t * NPER_ + n];
    int   c  = choice[t * NPER_ + n];
    int   nb = nnIdx[(size_t)t * KNN + c];
    const v4f* xs = (const v4f*)(X + (size_t)t  * D_N);
    const v4f* xn = (const v4f*)(X + (size_t)nb * D_N);
    v4f*       o  = (v4f*)(out + (size_t)r * D_N);
    for (int d4 = threadIdx.x; d4 < D_N / 4; d4 += blockDim.x) {
        v4f a = xs[d4], b = xn[d4];
        o[d4] = a + g * (b - a);
    }
}

// ---------------------------------------------------------------------------
extern "C" void kernel_launch(void* const* d_in, const int* in_sizes, int n_in,
                              void* d_out, int out_size, void* d_ws, size_t ws_size,
                              hipStream_t stream) {
    const float* X      = (const float*)d_in[0];   // [8192,512] fp32
    const float* gaps   = (const float*)d_in[1];   // [8192,4]   fp32
    const int*   choice = (const int*)d_in[2];     // [8192,4]   int32
    float*       out    = (float*)d_out;           // [32768,512] fp32

    float* sq    = (float*)d_ws;                               // 32 KB
    int*   nnIdx = (int*)((char*)d_ws + T_N * sizeof(float));  // 160 KB

    smote_norms   <<<T_N / 8,    256, 0, stream>>>(X, sq);
    smote_knn     <<<T_N / 16,   256, 0, stream>>>(X, sq, nnIdx);
    smote_populate<<<T_N * NPER_, 128, 0, stream>>>(X, gaps, choice, nnIdx, out);
}